// DualModeSinkhorn_4904852652235
// MI455X (gfx1250) — compile-verified
//
#include <hip/hip_runtime.h>
#include <stdint.h>

// DualModeSinkhorn on MI455X (gfx1250).
// Per-pixel independent 2x2 Sinkhorn normalization (20 iters), then exp.
// 302 MB traffic -> ~13 us floor @ 23.3 TB/s. Compute: ~480 packed-f32 adds
// per 512 elements -> stays under the roofline crossover.
//
// Data movement: 4 channels sit at stride H*W = 589824 floats; each wave's
// tile is a TDM 2-D descriptor (4 rows x 128 px, row stride HW) loaded with
// tensor_load_to_lds, double-buffered on TENSORcnt. Stage geometry is chosen
// so the global address advances by a compile-time constant per stage
// (tile index += 9216 = 2*4608  =>  batch += 2, spatial offset invariant),
// eliminating all per-stage division/multiply descriptor math.

typedef float f4 __attribute__((ext_vector_type(4)));
typedef unsigned int u32x4 __attribute__((ext_vector_type(4)));
typedef int i32x8 __attribute__((ext_vector_type(8)));
typedef int i32x4 __attribute__((ext_vector_type(4)));

#define SINK_HW       589824     // 768*768 (channel stride, elements)
#define TILE_PIX      128        // pixels per TDM tile (per wave per stage)
#define TILES_PER_B   4608       // SINK_HW / TILE_PIX
#define NBLOCKS       1152
#define WAVES_PB      8
#define NSTAGES       8          // 1152*8 waves * 8 stages * 128 px = all pixels
#define STAGE_STEP    (8LL * SINK_HW)   // +2 batches of 4*HW elements per stage

__device__ __forceinline__ void tdm_issue_tile(unsigned lds_addr,
                                               unsigned long long gaddr) {
    // D# group 0 (128b): count=1 | lds_addr | global_addr[56:0] | type=2
    u32x4 g0;
    g0[0] = 1u;                                   // count=1, user mode
    g0[1] = lds_addr;                             // LDS byte address
    g0[2] = (unsigned)gaddr;                      // global_addr[31:0] -> bits 95:64
    g0[3] = ((unsigned)(gaddr >> 32) & 0x01FFFFFFu) | (2u << 30); // [56:32], type=2

    // D# group 1 (256b): wg_mask=0, data_size=4B, 2-D tile dims/strides (constants)
    i32x8 g1;
    g1[0] = (int)(2u << 16);                 // data_size=2 (4 bytes); wg_mask=0
    g1[1] = (int)((unsigned)TILE_PIX << 16); // tensor_dim0[15:0] @ bits 63:48
    g1[2] = (int)(4u << 16);                 // tensor_dim1=4 (low half)
    g1[3] = (int)((unsigned)TILE_PIX << 16); // tile_dim0=128
    g1[4] = 4;                               // tile_dim1=4, tile_dim2=0
    g1[5] = SINK_HW;                         // tensor_dim0_stride[31:0] (elements)
    g1[6] = 0;
    g1[7] = 0;

    i32x4 z4 = {0, 0, 0, 0};                 // groups 2/3: unused (2-D tensor)
    i32x8 z8 = {0, 0, 0, 0, 0, 0, 0, 0};
    __builtin_amdgcn_tensor_load_to_lds(g0, g1, z4, z4, z8, 0);  // 6-arg form
}

__global__ __launch_bounds__(256)
void sinkhorn2x2_tdm_kernel(const float* __restrict__ in, float* __restrict__ out) {
    __shared__ float smem[WAVES_PB * 2 * 4 * TILE_PIX];  // 8 waves * 2 bufs * 2KB = 32KB

    const int wave = __builtin_amdgcn_readfirstlane((int)(threadIdx.x >> 5));
    const int lane = (int)(threadIdx.x & 31u);

    float* buf0 = &smem[(wave * 2 + 0) * (4 * TILE_PIX)];
    float* buf1 = &smem[(wave * 2 + 1) * (4 * TILE_PIX)];

    // Stage-0 tile index: t0 in [0, 9216) -> b0 = t0 >= 4608, sp stage-invariant.
    const long long t0 = (long long)blockIdx.x * WAVES_PB + wave;
    const long long b0 = (t0 >= TILES_PER_B) ? 1 : 0;
    const long long sp = (t0 - b0 * TILES_PER_B) * TILE_PIX;

    const float* gin  = in  + b0 * (4LL * SINK_HW) + sp;            // tile base, stage 0
    float*       gout = out + b0 * (4LL * SINK_HW) + sp + lane * 4; // lane's out base

    tdm_issue_tile((unsigned)(uintptr_t)buf0, (unsigned long long)(uintptr_t)gin);
    const float* gin_next = gin + STAGE_STEP;

    for (int s = 0; s < NSTAGES; ++s) {
        if (s + 1 < NSTAGES) {
            tdm_issue_tile((unsigned)(uintptr_t)(((s + 1) & 1) ? buf1 : buf0),
                           (unsigned long long)(uintptr_t)gin_next);
            gin_next += STAGE_STEP;
            __builtin_amdgcn_s_wait_tensorcnt(1);   // oldest (tile s) complete
        } else {
            __builtin_amdgcn_s_wait_tensorcnt(0);
        }
        asm volatile("" ::: "memory");              // keep LDS reads below the wait

        const float* bp = (s & 1) ? buf1 : buf0;

        // ds_load_b128 per channel: lane handles 4 consecutive pixels
        f4 a  = *(const f4*)(bp + 0 * TILE_PIX + lane * 4);
        f4 b2 = *(const f4*)(bp + 1 * TILE_PIX + lane * 4);
        f4 c  = *(const f4*)(bp + 2 * TILE_PIX + lane * 4);
        f4 d  = *(const f4*)(bp + 3 * TILE_PIX + lane * 4);

        // Faithful reference FP order: row-marginal subtract, col-marginal subtract.
#pragma unroll
        for (int it = 0; it < 20; ++it) {
            f4 r0 = a + b2;
            f4 r1 = c + d;
            a -= r0; b2 -= r0;
            c -= r1; d -= r1;
            f4 c0 = a + c;
            f4 c1 = b2 + d;
            a -= c0; c -= c0;
            b2 -= c1; d -= c1;
        }

        f4 ea, eb, ec, ed;
#pragma unroll
        for (int k = 0; k < 4; ++k) {
            ea[k] = __expf(a[k]);
            eb[k] = __expf(b2[k]);
            ec[k] = __expf(c[k]);
            ed[k] = __expf(d[k]);
        }

        // Non-temporal 128-bit stores (write-once stream, > L2 capacity)
        __builtin_nontemporal_store(ea, (f4*)(gout));
        __builtin_nontemporal_store(eb, (f4*)(gout + SINK_HW));
        __builtin_nontemporal_store(ec, (f4*)(gout + 2LL * SINK_HW));
        __builtin_nontemporal_store(ed, (f4*)(gout + 3LL * SINK_HW));
        gout += STAGE_STEP;

        asm volatile("" ::: "memory");  // don't sink next issue above compute
    }
}

extern "C" void kernel_launch(void* const* d_in, const int* in_sizes, int n_in,
                              void* d_out, int out_size, void* d_ws, size_t ws_size,
                              hipStream_t stream) {
    (void)in_sizes; (void)n_in; (void)out_size; (void)d_ws; (void)ws_size;
    const float* in  = (const float*)d_in[0];
    float*       out = (float*)d_out;

    dim3 grid(NBLOCKS), block(256);   // 1152 blocks * 8 waves * 8 stages * 128 px
    hipLaunchKernelGGL(sinkhorn2x2_tdm_kernel, grid, block, 0, stream, in, out);
}